// LBPKernel_83906481095327
// MI455X (gfx1250) — compile-verified
//
#include <hip/hip_runtime.h>
#include <cstdint>

#define HW (512 * 512)
#define TX 64
#define TY 32
// halo tile: 66 x 34 = 2244 gray elements
#define HALO_N 2244
// raw staging: rows of 72 floats (18 aligned float4 segs), 34 rows per channel
#define RAW_ROW 72
#define RAW_CH  (34 * RAW_ROW)   // 2448 floats per channel
#define NSEG    (3 * 34 * 18)    // 1836 b128 DMA ops per block

// ---- CDNA5 async global->LDS DMA (gfx1250) -------------------------------
__device__ __forceinline__ void async_g2l_b128(uint32_t lds_byte_off, const float* gaddr) {
    // VDST = VGPR holding LDS byte offset, VADDR = 64-bit global address (GV mode)
    asm volatile("global_load_async_to_lds_b128 %0, %1, off"
                 :: "v"(lds_byte_off), "v"(gaddr)
                 : "memory");
}
__device__ __forceinline__ void wait_asynccnt0() {
    asm volatile("s_wait_asynccnt 0" ::: "memory");
}

// ---- Kernel 0: zero per-image histograms ---------------------------------
__global__ __launch_bounds__(256) void zero_hist(unsigned int* __restrict__ ghist) {
    ghist[blockIdx.x * 256 + threadIdx.x] = 0u;
}

// ---- Kernel 1: gray + LBP codes + histogram ------------------------------
__global__ __launch_bounds__(256) void lbp_pass1(const float* __restrict__ img,
                                                 unsigned char* __restrict__ codes,
                                                 unsigned int* __restrict__ ghist) {
    __shared__ __align__(16) float s_raw[3 * RAW_CH];  // raw R,G,B halo tiles
    __shared__ float s_gray[HALO_N];                   // gray halo (66 wide, 34 tall)
    __shared__ unsigned int s_hist[256];

    const int tid = threadIdx.x;
    const int bx = blockIdx.x, by = blockIdx.y, n = blockIdx.z;
    const int x0 = bx * TX - 1, y0 = by * TY - 1;
    const int x0a = bx * TX - 4;                       // 16B-aligned row window start
    const float* imgN = img + (size_t)n * 3 * HW;
    const uint32_t raw_base = (uint32_t)(uintptr_t)(&s_raw[0]);

    // Async b128 DMA of three channel halo tiles into LDS (512B per wave op).
    for (int k = 0; k < 8; ++k) {
        const int q = tid + k * 256;
        if (q < NSEG) {
            const int c   = q / 612;            // 612 = 34*18 segs per channel
            const int rr  = q - c * 612;
            const int ly  = rr / 18;
            const int seg = rr - ly * 18;
            int gy = y0 + ly;   gy  = gy  < 0 ? 0 : (gy  > 511 ? 511 : gy);
            int gx4 = x0a + seg * 4;
            gx4 = gx4 < 0 ? 0 : (gx4 > 508 ? 508 : gx4);   // stays 16B aligned, in-row
            const float* gp = imgN + c * HW + (gy << 9) + gx4;
            async_g2l_b128(raw_base + (uint32_t)((c * RAW_CH + ly * RAW_ROW + seg * 4) * 4), gp);
        }
    }
    s_hist[tid] = 0u;           // overlap with DMA
    wait_asynccnt0();
    __syncthreads();

    // Gray conversion (no FMA contraction: match XLA's mul-then-add rounding).
    // Slot (ly, lx+3) in raw rows corresponds to gray element (ly, lx).
    for (int k = 0; k < 9; ++k) {
        const int idx = tid + k * 256;
        if (idx < HALO_N) {
            const int ly = idx / 66, lx = idx - ly * 66;
            const int gy = y0 + ly, gx = x0 + lx;
            const bool inb = (gy >= 0) & (gy < 512) & (gx >= 0) & (gx < 512);
            const int ri = ly * RAW_ROW + lx + 3;
            const float r = s_raw[ri];
            const float g = s_raw[RAW_CH + ri];
            const float b = s_raw[2 * RAW_CH + ri];
            float gr;
            {
#pragma clang fp contract(off)
                gr = r * 0.299f + g * 0.587f + b * 0.114f;
            }
            s_gray[idx] = inb ? gr : 0.0f;   // zero padding == conv SAME
        }
    }
    __syncthreads();

    // LBP codes + LDS histogram. neighbor = gray[py+r][px+c], center = gray[py+1][px+1].
    for (int k = 0; k < 8; ++k) {
        const int i  = tid + k * 256;
        const int py = i >> 6, px = i & 63;
        const int base = py * 66 + px;
        const float c = s_gray[base + 66 + 1];
        unsigned code = 0;
        code |= (s_gray[base + 0 * 66 + 2] >= c) ?   1u : 0u;  // (0,2)
        code |= (s_gray[base + 1 * 66 + 2] >= c) ?   2u : 0u;  // (1,2)
        code |= (s_gray[base + 2 * 66 + 2] >= c) ?   4u : 0u;  // (2,2)
        code |= (s_gray[base + 2 * 66 + 1] >= c) ?   8u : 0u;  // (2,1)
        code |= (s_gray[base + 2 * 66 + 0] >= c) ?  16u : 0u;  // (2,0)
        code |= (s_gray[base + 1 * 66 + 0] >= c) ?  32u : 0u;  // (1,0)
        code |= (s_gray[base + 0 * 66 + 0] >= c) ?  64u : 0u;  // (0,0)
        code |= (s_gray[base + 0 * 66 + 1] >= c) ? 128u : 0u;  // (0,1)
        codes[(size_t)n * HW + (size_t)(y0 + 1 + py) * 512 + (x0 + 1 + px)] =
            (unsigned char)code;
        atomicAdd(&s_hist[code], 1u);
    }
    __syncthreads();

    const unsigned cnt = s_hist[tid];
    if (cnt) atomicAdd(&ghist[n * 256 + tid], cnt);
}

// ---- Kernel 2: normalize hist + derive image stats from hist moments ----
__global__ __launch_bounds__(256) void lbp_stats(const unsigned int* __restrict__ ghist,
                                                 float* __restrict__ out_hist,
                                                 float* __restrict__ stats) {
    __shared__ double red[256];
    const int t = threadIdx.x, n = blockIdx.x;
    const float h = (float)ghist[n * 256 + t];

    // sum(h) (exact: integers)
    red[t] = (double)h; __syncthreads();
    for (int s = 128; s > 0; s >>= 1) { if (t < s) red[t] += red[t + s]; __syncthreads(); }
    const double sum = red[0]; __syncthreads();
    const float fmean = (float)(sum / 256.0);        // = 1024 exactly
    const float d = h - fmean;

    // sum(d^2) (exact: integers)
    red[t] = (double)d * (double)d; __syncthreads();
    for (int s = 128; s > 0; s >>= 1) { if (t < s) red[t] += red[t + s]; __syncthreads(); }
    const double sd2 = red[0]; __syncthreads();
    const float stdh = sqrtf((float)(sd2 / 255.0));  // ddof=1
    out_hist[n * 256 + t] = d / stdh;

    // pixel moments from histogram: S1 = sum v*h, S2 = sum v^2*h (exact)
    red[t] = (double)h * (double)t; __syncthreads();
    for (int s = 128; s > 0; s >>= 1) { if (t < s) red[t] += red[t + s]; __syncthreads(); }
    const double s1 = red[0]; __syncthreads();
    red[t] = (double)h * (double)(t * t); __syncthreads();
    for (int s = 128; s > 0; s >>= 1) { if (t < s) red[t] += red[t + s]; __syncthreads(); }
    if (t == 0) {
        const double s2 = red[0];
        const double N = (double)HW;
        const double meanP = s1 / N;
        const double varP = (s2 - s1 * s1 / N) / (N - 1.0);  // ddof=1
        stats[n * 2 + 0] = (float)meanP;
        stats[n * 2 + 1] = (float)sqrt(varP);
    }
}

// ---- Kernel 3: normalized LBP image via 256-entry LUT -------------------
__global__ __launch_bounds__(256) void lbp_norm(const unsigned char* __restrict__ codes,
                                                const float* __restrict__ stats,
                                                float* __restrict__ out_img) {
    __shared__ float lut[256];
    const int t = threadIdx.x, n = blockIdx.y;
    const float m = stats[n * 2 + 0], s = stats[n * 2 + 1];
    lut[t] = ((float)t - m) / s;
    __syncthreads();

    const int i4 = blockIdx.x * 256 + t;                 // 65536 uchar4 per image
    const uchar4 c = ((const uchar4*)(codes + (size_t)n * HW))[i4];
    const float4 o = make_float4(lut[c.x], lut[c.y], lut[c.z], lut[c.w]);
    ((float4*)(out_img + (size_t)n * HW))[i4] = o;
}

extern "C" void kernel_launch(void* const* d_in, const int* in_sizes, int n_in,
                              void* d_out, int out_size, void* d_ws, size_t ws_size,
                              hipStream_t stream) {
    (void)in_sizes; (void)n_in; (void)out_size; (void)ws_size;
    const float* img = (const float*)d_in[0];
    float* out_hist = (float*)d_out;                  // 32*256 floats
    float* out_img  = (float*)d_out + 32 * 256;       // 32*512*512 floats

    char* ws = (char*)d_ws;
    unsigned int* ghist   = (unsigned int*)ws;            // 32*256 u32  @ 0
    float*        stats   = (float*)(ws + 32768);         // 64 floats   @ 32768
    unsigned char* codes  = (unsigned char*)(ws + 65536); // 8 MB codes  @ 65536

    zero_hist<<<dim3(32), dim3(256), 0, stream>>>(ghist);
    lbp_pass1<<<dim3(512 / TX, 512 / TY, 32), dim3(256), 0, stream>>>(img, codes, ghist);
    lbp_stats<<<dim3(32), dim3(256), 0, stream>>>(ghist, out_hist, stats);
    lbp_norm<<<dim3(256, 32), dim3(256), 0, stream>>>(codes, stats, out_img);
}